// GCNEncoder_48876727828952
// MI455X (gfx1250) — compile-verified
//
#include <hip/hip_runtime.h>
#include <hip/hip_bf16.h>

// GCN encoder, 2 layers: h = relu(Agg(x@W1)+b1); out = relu(Agg(h@W2)+b2)
// Agg(z)[c] = dinv[c]^2*z[c] + sum_e norm[e]*z[row[e]]  (self-loop folded into init)
// GEMMs use exact-fp32 V_WMMA_F32_16X16X4_F32 (16x16 tile per wave, K stepped by 4).
// Aggregation is L2-resident (xw = 25.6MB << 192MB L2): one lane per (edge,feature)
// so each edge's 64/32-float gather+atomic burst is contiguous.

typedef __attribute__((ext_vector_type(2))) float v2f;
typedef __attribute__((ext_vector_type(8))) float v8f;

#define TFI 64   // F_IN
#define TFH 64   // F_HID
#define TFL 32   // F_LAT

// ---------------- degree / norm pipeline ----------------

__global__ void k_fill(float* __restrict__ p, int n, float v) {
  int i = blockIdx.x * blockDim.x + threadIdx.x;
  if (i < n) p[i] = v;
}

__global__ void k_degree(const int* __restrict__ col, const float* __restrict__ w,
                         float* __restrict__ deg, int E) {
  int e = blockIdx.x * blockDim.x + threadIdx.x;
  if (e < E) atomicAdd(&deg[col[e]], w[e]);
}

__global__ void k_rsqrt_inplace(float* __restrict__ d, int n) {
  int i = blockIdx.x * blockDim.x + threadIdx.x;
  if (i < n) {
    float v = d[i];
    d[i] = (v > 0.0f) ? rsqrtf(v) : 0.0f;
  }
}

__global__ void k_norm(const int* __restrict__ row, const int* __restrict__ col,
                       const float* __restrict__ w, const float* __restrict__ dinv,
                       float* __restrict__ nrm, int E) {
  int e = blockIdx.x * blockDim.x + threadIdx.x;
  if (e < E) nrm[e] = dinv[row[e]] * w[e] * dinv[col[e]];
}

// ---------------- dense GEMM via fp32 WMMA ----------------
// One wave computes one 16x16 tile of Out = A[MxK] @ W[KxNout].
// A-operand (16x4 f32, ISA 7.12.2): lane<16 holds {K=k0,k0+1} for row M=lane,
// lane>=16 holds {K=k0+2,k0+3}. B striped the same way across K.
// C/D: VGPR j -> (M = j + 8*(lane>=16), N = lane&15).

__global__ void k_gemm_wmma(const float* __restrict__ A, const float* __restrict__ W,
                            float* __restrict__ Out, int M, int K, int Nout) {
  int wavesPerBlock = blockDim.x >> 5;
  int wave = blockIdx.x * wavesPerBlock + (threadIdx.x >> 5);
  int tilesN = Nout >> 4;
  int tileM = wave / tilesN;
  int tileN = wave - tileM * tilesN;
  if (tileM * 16 >= M) return;          // whole wave exits together: EXEC stays all-1s

  int lane = threadIdx.x & 31;
  int half = lane >> 4;                 // 0: K pair {0,1}; 1: K pair {2,3}
  int l    = lane & 15;

  const float* arow = A + (size_t)(tileM * 16 + l) * K;
  const float* bcol = W + (size_t)(tileN * 16 + l);

  v8f c = {};
  for (int k0 = 0; k0 < K; k0 += 4) {
    v2f a, b;
    a.x = arow[k0 + 2 * half + 0];
    a.y = arow[k0 + 2 * half + 1];
    b.x = bcol[(size_t)(k0 + 2 * half + 0) * Nout];
    b.y = bcol[(size_t)(k0 + 2 * half + 1) * Nout];
    c = __builtin_amdgcn_wmma_f32_16x16x4_f32(false, a, false, b, (short)0, c,
                                              false, false);
  }

  float* orow = Out + (size_t)tileN * 16 + l;
  int baseM = tileM * 16 + half * 8;
#pragma unroll
  for (int j = 0; j < 8; ++j)
    orow[(size_t)(baseM + j) * Nout] = c[j];
}

// ---------------- aggregation ----------------

template <int F>
__global__ void k_agg_init(const float* __restrict__ xw, const float* __restrict__ dinv,
                           float* __restrict__ out, int N) {
  constexpr int SH = (F == 64) ? 6 : 5;
  long long idx = (long long)blockIdx.x * blockDim.x + threadIdx.x;
  if (idx >= (long long)N * F) return;
  int i = (int)(idx >> SH);
  float di = dinv[i];
  out[idx] = di * di * xw[idx];        // self-loop term; also un-poisons the buffer
}

template <int F>
__global__ void k_agg_edges(const int* __restrict__ row, const int* __restrict__ col,
                            const float* __restrict__ nrm, const float* __restrict__ xw,
                            float* __restrict__ out, int E) {
  constexpr int SH = (F == 64) ? 6 : 5;
  long long idx = (long long)blockIdx.x * blockDim.x + threadIdx.x;
  if (idx >= (long long)E * F) return;
  int e = (int)(idx >> SH);
  int f = (int)(idx & (F - 1));
  int r = row[e];
  int c = col[e];
  atomicAdd(&out[(size_t)c * F + f], nrm[e] * xw[(size_t)r * F + f]);
}

template <int F>
__global__ void k_bias_relu(float* __restrict__ p, const float* __restrict__ b, int N) {
  long long idx = (long long)blockIdx.x * blockDim.x + threadIdx.x;
  if (idx >= (long long)N * F) return;
  int f = (int)(idx & (F - 1));
  float v = p[idx] + b[f];
  p[idx] = v > 0.0f ? v : 0.0f;
}

// ---------------- launcher ----------------

extern "C" void kernel_launch(void* const* d_in, const int* in_sizes, int n_in,
                              void* d_out, int out_size, void* d_ws, size_t ws_size,
                              hipStream_t stream) {
  const float* x  = (const float*)d_in[0];   // [N, 64]
  const int*   ei = (const int*)d_in[1];     // [2, E] flat: rows then cols
  const float* w  = (const float*)d_in[2];   // [E]
  const float* W1 = (const float*)d_in[3];   // [64, 64]
  const float* b1 = (const float*)d_in[4];   // [64]
  const float* W2 = (const float*)d_in[5];   // [64, 32]
  const float* b2 = (const float*)d_in[6];   // [32]
  float* out = (float*)d_out;                // [N, 32]

  const int N = in_sizes[0] / TFI;           // 100000 (divisible by 16)
  const int E = in_sizes[2];                 // 1600000
  const int* row = ei;
  const int* col = ei + E;

  // workspace: dinv[N] | norm[E] | bufA[N*64] | bufB[N*64]  (~58 MB fp32)
  float* ws   = (float*)d_ws;
  float* dinv = ws;
  float* nrm  = dinv + N;
  float* bufA = nrm + E;
  float* bufB = bufA + (size_t)N * TFH;

  const int T = 256;
  auto cdiv = [](long long a, long long b) { return (int)((a + b - 1) / b); };

  // normalization coefficients
  k_fill<<<cdiv(N, T), T, 0, stream>>>(dinv, N, 1.0f);            // self-loop weight
  k_degree<<<cdiv(E, T), T, 0, stream>>>(col, w, dinv, E);
  k_rsqrt_inplace<<<cdiv(N, T), T, 0, stream>>>(dinv, N);
  k_norm<<<cdiv(E, T), T, 0, stream>>>(row, col, w, dinv, nrm, E);

  // ---- layer 1: bufB = relu(Agg(x@W1) + b1) ----
  {
    int tiles = (N / 16) * (TFH / 16);                            // 25000 waves
    k_gemm_wmma<<<cdiv(tiles, 8), 256, 0, stream>>>(x, W1, bufA, N, TFI, TFH);
    k_agg_init<TFH><<<cdiv((long long)N * TFH, T), T, 0, stream>>>(bufA, dinv, bufB, N);
    k_agg_edges<TFH><<<cdiv((long long)E * TFH, T), T, 0, stream>>>(row, col, nrm, bufA, bufB, E);
    k_bias_relu<TFH><<<cdiv((long long)N * TFH, T), T, 0, stream>>>(bufB, b1, N);
  }

  // ---- layer 2: out = relu(Agg(bufB@W2) + b2) ----
  {
    int tiles = (N / 16) * (TFL / 16);                            // 12500 waves
    k_gemm_wmma<<<cdiv(tiles, 8), 256, 0, stream>>>(bufB, W2, bufA, N, TFH, TFL);
    k_agg_init<TFL><<<cdiv((long long)N * TFL, T), T, 0, stream>>>(bufA, dinv, out, N);
    k_agg_edges<TFL><<<cdiv((long long)E * TFL, T), T, 0, stream>>>(row, col, nrm, bufA, out, E);
    k_bias_relu<TFL><<<cdiv((long long)N * TFL, T), T, 0, stream>>>(out, b2, N);
  }
}